// RegisterFlowAttention_61211873903116
// MI455X (gfx1250) — compile-verified
//
#include <hip/hip_runtime.h>
#include <hip/hip_bf16.h>
#include <math.h>

#define SEQ   4096
#define DM    256
#define NH    8
#define HD    32
#define QKVD  768
#define MASKW 128   // SEQ/32 words per row

typedef __attribute__((ext_vector_type(2))) float v2f;
typedef __attribute__((ext_vector_type(8))) float v8f;

__device__ __forceinline__ v8f wmma_f32_4(v2f a, v2f b, v8f c) {
  // D = A(16x4) * B(4x16) + C(16x16), fp32, wave32
  return __builtin_amdgcn_wmma_f32_16x16x4_f32(false, a, false, b, (short)0, c, false, false);
}

// ---------------- mask construction ----------------
__global__ void zero_u32_kernel(unsigned* p, int n) {
  int i = blockIdx.x * blockDim.x + threadIdx.x;
  if (i < n) p[i] = 0u;
}

__global__ void set_edges_kernel(const int* __restrict__ edges, int E, unsigned* __restrict__ mask) {
  int i = blockIdx.x * blockDim.x + threadIdx.x;
  if (i >= E) return;
  int a = edges[2 * i + 0];
  int b = edges[2 * i + 1];
  atomicOr(&mask[(size_t)a * MASKW + (b >> 5)], 1u << (b & 31));
  atomicOr(&mask[(size_t)b * MASKW + (a >> 5)], 1u << (a & 31));
}

// ---------------- generic C = A @ W^T + bias via f32 WMMA ----------------
// one wave per 16x16 output tile; A:[M,K] row-major, W:[N,K] row-major
__global__ void gemm_xwT_wmma(const float* __restrict__ A, const float* __restrict__ W,
                              const float* __restrict__ bias, float* __restrict__ C,
                              int M, int N, int K) {
  int wave = (blockIdx.x * blockDim.x + threadIdx.x) >> 5;
  int lane = threadIdx.x & 31;
  int hf = lane >> 4;        // half of wave: K-pair select
  int l  = lane & 15;        // M (for A) / N (for B,C)
  int tilesN = N >> 4;
  int tm = wave / tilesN;
  int tn = wave % tilesN;
  const float* arow = A + (size_t)(tm * 16 + l) * K;
  const float* wrow = W + (size_t)(tn * 16 + l) * K;
  v8f acc = {};
  #pragma unroll 4
  for (int k = 0; k < K; k += 4) {
    v2f a, b;
    a.x = arow[k + 2 * hf];     a.y = arow[k + 2 * hf + 1];
    b.x = wrow[k + 2 * hf];     b.y = wrow[k + 2 * hf + 1];
    acc = wmma_f32_4(a, b, acc);
  }
  float bn = bias[tn * 16 + l];
  #pragma unroll
  for (int i = 0; i < 8; i++) {
    int m = tm * 16 + i + 8 * hf;
    C[(size_t)m * N + tn * 16 + l] = acc[i] + bn;
  }
}

// ---------------- pass A: per-(head, q-tile) online row max / sumexp ----------------
__global__ void attn_stats_kernel(const float* __restrict__ qkv, const unsigned* __restrict__ mask,
                                  float* __restrict__ mbuf, float* __restrict__ lbuf) {
  int wave = (blockIdx.x * blockDim.x + threadIdx.x) >> 5;
  int lane = threadIdx.x & 31;
  int hf = lane >> 4, l = lane & 15;
  int head = wave & (NH - 1);
  int qt   = wave >> 3;
  int q0 = qt * 16;
  const float NEG_INF = -__builtin_inff();
  const float SCALE = 0.17677669529663687f; // 1/sqrt(32)
  const float* qrow = qkv + (size_t)(q0 + l) * QKVD + head * HD;
  float mr[8], lr[8];
  #pragma unroll
  for (int i = 0; i < 8; i++) { mr[i] = NEG_INF; lr[i] = 0.f; }

  for (int kt = 0; kt < SEQ / 16; kt++) {
    int k0 = kt * 16;
    const float* krow = qkv + (size_t)(k0 + l) * QKVD + DM + head * HD;
    v8f acc = {};
    #pragma unroll
    for (int kk = 0; kk < HD; kk += 4) {
      v2f a, b;
      a.x = qrow[kk + 2 * hf]; a.y = qrow[kk + 2 * hf + 1];
      b.x = krow[kk + 2 * hf]; b.y = krow[kk + 2 * hf + 1];
      acc = wmma_f32_4(a, b, acc);
    }
    int wword = k0 >> 5;
    int wsh   = (k0 & 31) + l;
    #pragma unroll
    for (int i = 0; i < 8; i++) {
      int m = q0 + i + 8 * hf;
      unsigned bits = mask[(size_t)m * MASKW + wword];
      float s = ((bits >> wsh) & 1u) ? acc[i] * SCALE : NEG_INF;
      float rm = s;
      rm = fmaxf(rm, __shfl_xor(rm, 1, 16));
      rm = fmaxf(rm, __shfl_xor(rm, 2, 16));
      rm = fmaxf(rm, __shfl_xor(rm, 4, 16));
      rm = fmaxf(rm, __shfl_xor(rm, 8, 16));
      float mnew = fmaxf(mr[i], rm);
      if (mnew > NEG_INF) {  // uniform within each 16-lane group
        float fac = (mr[i] > NEG_INF) ? __expf(mr[i] - mnew) : 0.f;
        float e   = (s > NEG_INF) ? __expf(s - mnew) : 0.f;
        e += __shfl_xor(e, 1, 16);
        e += __shfl_xor(e, 2, 16);
        e += __shfl_xor(e, 4, 16);
        e += __shfl_xor(e, 8, 16);
        lr[i] = lr[i] * fac + e;
        mr[i] = mnew;
      }
    }
  }
  if (l == 0) {
    #pragma unroll
    for (int i = 0; i < 8; i++) {
      int m = q0 + i + 8 * hf;
      mbuf[head * SEQ + m] = mr[i];
      lbuf[head * SEQ + m] = lr[i];
    }
  }
}

// ---------------- pass B: probs, averaged probs, ctx = P @ V ----------------
// one block per 16-row q-tile; wave w == head w
__global__ void attn_ctx_kernel(const float* __restrict__ qkv, const unsigned* __restrict__ mask,
                                const float* __restrict__ mbuf, const float* __restrict__ lbuf,
                                float* __restrict__ ctx, float* __restrict__ attnw) {
  __shared__ float avg[256];          // 16x16 head-averaged prob tile
  __shared__ float pbuf[NH][16][16];  // per-wave prob staging for C->A relayout
  int t = threadIdx.x;
  int wave = t >> 5, lane = t & 31;
  int hf = lane >> 4, l = lane & 15;
  int head = wave;
  int q0 = blockIdx.x * 16;
  const float SCALE = 0.17677669529663687f;
  const float* qrow = qkv + (size_t)(q0 + l) * QKVD + head * HD;

  float mr[8], invl[8];
  #pragma unroll
  for (int i = 0; i < 8; i++) {
    int m = q0 + i + 8 * hf;
    mr[i] = mbuf[head * SEQ + m];
    float lv = lbuf[head * SEQ + m];
    invl[i] = (lv > 0.f) ? 1.f / lv : 0.f;
  }
  avg[t] = 0.f;
  v8f cacc0 = {}, cacc1 = {};
  __syncthreads();

  for (int kt = 0; kt < SEQ / 16; kt++) {
    int k0 = kt * 16;
    const float* krow = qkv + (size_t)(k0 + l) * QKVD + DM + head * HD;
    v8f acc = {};
    #pragma unroll
    for (int kk = 0; kk < HD; kk += 4) {
      v2f a, b;
      a.x = qrow[kk + 2 * hf]; a.y = qrow[kk + 2 * hf + 1];
      b.x = krow[kk + 2 * hf]; b.y = krow[kk + 2 * hf + 1];
      acc = wmma_f32_4(a, b, acc);
    }
    int wword = k0 >> 5;
    int wsh   = (k0 & 31) + l;
    #pragma unroll
    for (int i = 0; i < 8; i++) {
      int m = q0 + i + 8 * hf;
      unsigned bits = mask[(size_t)m * MASKW + wword];
      float p = 0.f;
      if ((bits >> wsh) & 1u)
        p = __expf(acc[i] * SCALE - mr[i]) * invl[i];
      pbuf[wave][i + 8 * hf][l] = p;
      atomicAdd(&avg[(i + 8 * hf) * 16 + l], p * 0.125f);  // ds_add_f32
    }
    __syncthreads();  // avg atomics + pbuf writes visible

    // ctx += P(16x16) @ V(16x32): relayout P (C-layout -> A-layout) via LDS
    #pragma unroll
    for (int kk = 0; kk < 4; kk++) {
      v2f a;
      a.x = pbuf[wave][l][kk * 4 + 2 * hf];
      a.y = pbuf[wave][l][kk * 4 + 2 * hf + 1];
      const float* v0 = qkv + (size_t)(k0 + kk * 4 + 2 * hf)     * QKVD + 2 * DM + head * HD;
      const float* v1 = qkv + (size_t)(k0 + kk * 4 + 2 * hf + 1) * QKVD + 2 * DM + head * HD;
      v2f b0, b1;
      b0.x = v0[l];      b0.y = v1[l];       // d columns 0..15
      b1.x = v0[16 + l]; b1.y = v1[16 + l];  // d columns 16..31
      cacc0 = wmma_f32_4(a, b0, cacc0);
      cacc1 = wmma_f32_4(a, b1, cacc1);
    }

    // flush head-averaged prob tile and reset for next k-tile
    attnw[(size_t)(q0 + (t >> 4)) * SEQ + k0 + (t & 15)] = avg[t];
    avg[t] = 0.f;
    __syncthreads();
  }

  #pragma unroll
  for (int i = 0; i < 8; i++) {
    int m = q0 + i + 8 * hf;
    ctx[(size_t)m * DM + head * HD + l]      = cacc0[i];
    ctx[(size_t)m * DM + head * HD + 16 + l] = cacc1[i];
  }
}

extern "C" void kernel_launch(void* const* d_in, const int* in_sizes, int n_in,
                              void* d_out, int out_size, void* d_ws, size_t ws_size,
                              hipStream_t stream) {
  const float* x     = (const float*)d_in[0];
  const float* W_in  = (const float*)d_in[1];
  const float* b_in  = (const float*)d_in[2];
  const float* W_out = (const float*)d_in[3];
  const float* b_out = (const float*)d_in[4];
  const int*   edges = (const int*)d_in[5];
  int E = in_sizes[5] / 2;

  float* out   = (float*)d_out;             // [4096,256]
  float* attnw = out + (size_t)SEQ * DM;    // [4096,4096]

  // workspace carve-up (~18.3 MB)
  char* ws = (char*)d_ws;
  unsigned* mask = (unsigned*)ws;  ws += (size_t)SEQ * MASKW * sizeof(unsigned);
  float* qkv  = (float*)ws;        ws += (size_t)SEQ * QKVD * sizeof(float);
  float* mbuf = (float*)ws;        ws += (size_t)NH * SEQ * sizeof(float);
  float* lbuf = (float*)ws;        ws += (size_t)NH * SEQ * sizeof(float);
  float* ctx  = (float*)ws;        ws += (size_t)SEQ * DM * sizeof(float);

  int nmask = SEQ * MASKW;
  zero_u32_kernel<<<(nmask + 255) / 256, 256, 0, stream>>>(mask, nmask);
  set_edges_kernel<<<(E + 255) / 256, 256, 0, stream>>>(edges, E, mask);

  // qkv = x @ W_in^T + b_in : M=4096, N=768, K=256 -> 12288 waves / 8 per block
  gemm_xwT_wmma<<<(256 * 48) / 8, 256, 0, stream>>>(x, W_in, b_in, qkv, SEQ, QKVD, DM);

  // pass A: 8 heads x 256 q-tiles = 2048 waves
  attn_stats_kernel<<<(NH * (SEQ / 16)) / 8, 256, 0, stream>>>(qkv, mask, mbuf, lbuf);

  // pass B: one block per q-tile
  attn_ctx_kernel<<<SEQ / 16, 256, 0, stream>>>(qkv, mask, mbuf, lbuf, ctx, attnw);

  // out = ctx @ W_out^T + b_out : M=4096, N=256, K=256 -> 4096 waves
  gemm_xwT_wmma<<<(256 * 16) / 8, 256, 0, stream>>>(ctx, W_out, b_out, out, SEQ, DM, DM);
}